// YOLOLayer_25486335935114
// MI455X (gfx1250) — compile-verified
//
#include <hip/hip_runtime.h>
#include <cstdint>

// Problem constants (from the reference: bs=32, nA=3, nC=80, nG=76)
#define NG     76
#define NA     3
#define NCH    85                 // nC + 5
#define QROW   (NG / 4)           // 19 float4 per x-row (76 floats = 304B, 16B-aligned)
#define TILE_Q (NCH * QROW)       // 1615 float4 per (b,a,y) tile
#define TILE_E (NCH * NG)         // 6460 floats per tile (25840 B, 16B-aligned stride)

typedef float f4 __attribute__((ext_vector_type(4)));

#if defined(__gfx1250__) && __has_builtin(__builtin_amdgcn_tensor_load_to_lds)
#define TDM_PATH 1
typedef unsigned int u32x4 __attribute__((ext_vector_type(4)));
typedef int          i32x8 __attribute__((ext_vector_type(8)));
typedef int          i32x4 __attribute__((ext_vector_type(4)));
#elif defined(__gfx1250__) && __has_builtin(__builtin_amdgcn_global_load_async_to_lds_b128)
#define ASYNC_PATH 1
#endif

__device__ __forceinline__ float fast_sigmoid(float v) {
    // 1 / (1 + 2^(-v*log2e)) -> v_exp_f32 + v_rcp_f32
    return __builtin_amdgcn_rcpf(1.0f + __builtin_amdgcn_exp2f(v * -1.4426950408889634f));
}
__device__ __forceinline__ float fast_exp(float v) {
    return __builtin_amdgcn_exp2f(v * 1.4426950408889634f);
}

extern "C" __global__ void __launch_bounds__(256)
yolo_decode(const float* __restrict__ p,
            const float* __restrict__ anchors,
            const int*   __restrict__ img_size,
            float*       __restrict__ out)
{
    __shared__ __align__(16) float lds_raw[TILE_E]; // raw tile, [c][x]
    __shared__ __align__(16) float lds_t[TILE_E];   // transformed tile, [x][c]

    const int r  = blockIdx.x;      // r = (b*NA + a)*NG + y
    const int y  = r % NG;
    const int ba = r / NG;          // b*NA + a
    const int a  = ba % NA;

    const float* tile_src = p + (size_t)ba * NCH * (NG * NG) + (size_t)y * NG;

    const float stride = (float)(*img_size) * (1.0f / (float)NG);
    const float ax = anchors[2 * a + 0];   // exp(p)*anchor_vec*stride == exp(p)*anchor
    const float ay = anchors[2 * a + 1];

    // ---- Stage A: stage raw tile (85 rows x 76 f32, row stride 5776 f32) into LDS ----
#if TDM_PATH
    if (threadIdx.x < 32) {  // one wave issues the tensor DMA (EXEC ignored, per-wave issue)
        const unsigned           lds_addr = (unsigned)(uintptr_t)lds_raw;
        const unsigned long long ga       = (unsigned long long)(uintptr_t)tile_src;
        // D# group 0: count=1 | lds_addr | global_addr[56:0] | type=2
        u32x4 g0 = { 1u,
                     lds_addr,
                     (unsigned)ga,
                     (unsigned)((ga >> 32) & 0x1FFFFFFu) | (2u << 30) };
        // D# group 1: data_size=4B; tensor_dim0=76; tensor_dim1=85;
        //             tile_dim0=76; tile_dim1=85; tensor_dim0_stride=5776
        i32x8 g1 = { (int)(2u << 16),
                     (int)(NG  << 16),
                     (int)(NCH << 16),
                     (int)(NG  << 16),
                     (int)NCH,
                     (int)(NG * NG),
                     0, 0 };
        i32x4 g2 = { 0, 0, 0, 0 };
        i32x4 g3 = { 0, 0, 0, 0 };
        i32x8 g4 = { 0, 0, 0, 0, 0, 0, 0, 0 };   // extra group on this toolchain (zero-filled)
        __builtin_amdgcn_tensor_load_to_lds(g0, g1, g2, g3, g4, 0);
#if __has_builtin(__builtin_amdgcn_s_wait_tensorcnt)
        __builtin_amdgcn_s_wait_tensorcnt(0);
#else
        asm volatile("s_wait_tensorcnt 0" ::: "memory");
#endif
    }
#elif ASYNC_PATH
    for (int i = threadIdx.x; i < TILE_Q; i += 256) {
        const int c = i / QROW, q = i - c * QROW;
        const float* g = tile_src + (size_t)c * (NG * NG) + q * 4;
        typedef const __attribute__((address_space(1))) void* gp_t;
        typedef __attribute__((address_space(3))) void*       lp_t;
        gp_t gp = (gp_t)(uintptr_t)g;
        lp_t lp = (lp_t)(unsigned)(uintptr_t)(lds_raw + i * 4);
        __builtin_amdgcn_global_load_async_to_lds_b128(gp, lp, 0, 0);
    }
#if __has_builtin(__builtin_amdgcn_s_wait_asynccnt)
    __builtin_amdgcn_s_wait_asynccnt(0);
#else
    asm volatile("s_wait_asynccnt 0" ::: "memory");
#endif
#else
    for (int i = threadIdx.x; i < TILE_Q; i += 256) {
        const int c = i / QROW, q = i - c * QROW;
        f4 v = __builtin_nontemporal_load((const f4*)(tile_src + (size_t)c * (NG * NG)) + q);
        ((f4*)lds_raw)[i] = v;
    }
#endif
    __syncthreads();

    // ---- Stage B: transform + transpose [c][x] -> [x][c] in LDS ----
    for (int i = threadIdx.x; i < TILE_Q; i += 256) {
        const int c  = i / QROW, q = i - c * QROW;
        const int x0 = q * 4;
        f4 v = ((const f4*)lds_raw)[i];
        if (c == 0) {
            v.x = (fast_sigmoid(v.x) + (float)(x0 + 0)) * stride;
            v.y = (fast_sigmoid(v.y) + (float)(x0 + 1)) * stride;
            v.z = (fast_sigmoid(v.z) + (float)(x0 + 2)) * stride;
            v.w = (fast_sigmoid(v.w) + (float)(x0 + 3)) * stride;
        } else if (c == 1) {
            const float gy = (float)y;
            v.x = (fast_sigmoid(v.x) + gy) * stride;
            v.y = (fast_sigmoid(v.y) + gy) * stride;
            v.z = (fast_sigmoid(v.z) + gy) * stride;
            v.w = (fast_sigmoid(v.w) + gy) * stride;
        } else if (c == 2) {
            v.x = fast_exp(v.x) * ax;  v.y = fast_exp(v.y) * ax;
            v.z = fast_exp(v.z) * ax;  v.w = fast_exp(v.w) * ax;
        } else if (c == 3) {
            v.x = fast_exp(v.x) * ay;  v.y = fast_exp(v.y) * ay;
            v.z = fast_exp(v.z) * ay;  v.w = fast_exp(v.w) * ay;
        } else if (c == 4) {
            v.x = fast_sigmoid(v.x);   v.y = fast_sigmoid(v.y);
            v.z = fast_sigmoid(v.z);   v.w = fast_sigmoid(v.w);
        }
        // stride-85 word writes: gcd(85,64)=1 -> bank-conflict-free
        lds_t[(x0 + 0) * NCH + c] = v.x;
        lds_t[(x0 + 1) * NCH + c] = v.y;
        lds_t[(x0 + 2) * NCH + c] = v.z;
        lds_t[(x0 + 3) * NCH + c] = v.w;
    }
    __syncthreads();

    // ---- Stage C: coalesced 128-bit non-temporal streaming stores ----
    f4* __restrict__ o4  = (f4*)(out + (size_t)r * TILE_E);  // r*25840B, 16B-aligned
    const f4*        lt4 = (const f4*)lds_t;
    for (int j = threadIdx.x; j < TILE_Q; j += 256) {
        __builtin_nontemporal_store(lt4[j], o4 + j);
    }
}

extern "C" void kernel_launch(void* const* d_in, const int* in_sizes, int n_in,
                              void* d_out, int out_size, void* d_ws, size_t ws_size,
                              hipStream_t stream)
{
    const float* p       = (const float*)d_in[0];
    const float* anchors = (const float*)d_in[1];
    const int*   imgsz   = (const int*)d_in[2];
    float*       out     = (float*)d_out;

    const int bs   = in_sizes[0] / (NA * NCH * NG * NG);  // 32
    const int rows = bs * NA * NG;                        // 7296 blocks
    hipLaunchKernelGGL(yolo_decode, dim3(rows), dim3(256), 0, stream,
                       p, anchors, imgsz, out);
}